// QuantumPathway_84327387890416
// MI455X (gfx1250) — compile-verified
//
#include <hip/hip_runtime.h>
#include <hip/hip_bf16.h>
#include <math.h>

namespace {
constexpr int kNQ  = 12;
constexpr int kDim = 1 << kNQ;     // 4096 amplitudes
constexpr int kNL  = 6;
constexpr int kTPB = 256;          // 8 x wave32
constexpr int kAPT = kDim / kTPB;  // 16 amplitudes per thread
constexpr int kTilesPerWave = 2;   // 16 column tiles over 8 waves

typedef float v2f __attribute__((ext_vector_type(2)));
typedef float v8f __attribute__((ext_vector_type(8)));

// Group g acts on flat bits [4g, 4g+3]; row = nibble value, c = other 8 bits.
__device__ __forceinline__ int state_index(int g, int row, int c) {
  switch (g) {
    case 0:  return row | (c << 4);
    case 1:  return (c & 15) | (row << 4) | ((c >> 4) << 8);
    default: return (c & 15) | (((c >> 4) & 15) << 4) | (row << 8);
  }
}
}  // namespace

__global__ __launch_bounds__(kTPB) void qnn_vqc_kernel(
    const float* __restrict__ x, const float* __restrict__ params,
    const float* __restrict__ lin_w, const float* __restrict__ lin_b,
    float* __restrict__ out) {
  __shared__ float s_re[kDim];    // 16 KB
  __shared__ float s_im[kDim];    // 16 KB
  __shared__ float s_gate[256];   // 1 KB: 16x16 group gate / reduction scratch

  const int tid  = threadIdx.x;
  const int lane = tid & 31;
  const int wv   = tid >> 5;
  const int b    = blockIdx.x;

  // ---- Data encoding: prod_i RY(tanh(x_i)*pi)|0..0> is a real product state.
  // Flat bit p corresponds to qubit (11 - p) after the row-major reshape.
  float ec[kNQ], es[kNQ];
#pragma unroll
  for (int p = 0; p < kNQ; ++p) {
    float h = tanhf(x[b * kNQ + (kNQ - 1 - p)]) * 1.57079632679489662f;
    ec[p] = __cosf(h);
    es[p] = __sinf(h);
  }
#pragma unroll
  for (int r = 0; r < kAPT; ++r) {
    int k = r * kTPB + tid;
    float a = 1.0f;
#pragma unroll
    for (int p = 0; p < kNQ; ++p) a *= ((k >> p) & 1) ? es[p] : ec[p];
    s_re[k] = a;
    s_im[k] = 0.0f;
  }
  __syncthreads();

  for (int L = 0; L < kNL; ++L) {
    const float* pL = params + L * (2 * kNQ);

    // ---- RY layer: three 16x16 tensor-product gates via f32 WMMA ----
#pragma unroll
    for (int g = 0; g < 3; ++g) {
      // Build group gate G[m][m'] = prod_j RY(4g+j)[bit_j(m)][bit_j(m')].
      {
        int m = tid >> 4, mp = tid & 15;
        float v = 1.0f;
#pragma unroll
        for (int j = 0; j < 4; ++j) {
          int p = 4 * g + j;                 // flat bit -> qubit (11-p)
          float t = pL[kNQ - 1 - p] * 0.5f;
          float ct = __cosf(t), st = __sinf(t);
          int bm = (m >> j) & 1, bp = (mp >> j) & 1;
          v *= bm ? (bp ? ct : st) : (bp ? -st : ct);
        }
        s_gate[tid] = v;
      }
      __syncthreads();

      // Each wave owns 2 column tiles: D(16x16) = G @ S(16x256) per tile,
      // K=16 via 4 chained v_wmma_f32_16x16x4_f32, re/im planes separately.
      const int n    = lane & 15;
      const int half = lane >> 4;
      v8f cre[kTilesPerWave], cim[kTilesPerWave];
#pragma unroll
      for (int t = 0; t < kTilesPerWave; ++t) {
        cre[t] = v8f{0.f, 0.f, 0.f, 0.f, 0.f, 0.f, 0.f, 0.f};
        cim[t] = cre[t];
      }
#pragma unroll
      for (int kc = 0; kc < 4; ++kc) {
        const int k0 = kc * 4 + half * 2;
        v2f a;  // A frag: lanes 0-15 K=2h, second VGPR K=2h+1
        a.x = s_gate[n * 16 + k0];
        a.y = s_gate[n * 16 + k0 + 1];
#pragma unroll
        for (int t = 0; t < kTilesPerWave; ++t) {
          const int c  = (wv * kTilesPerWave + t) * 16 + n;
          const int i0 = state_index(g, k0, c);
          const int i1 = state_index(g, k0 + 1, c);
          v2f bre, bim;
          bre.x = s_re[i0]; bre.y = s_re[i1];
          bim.x = s_im[i0]; bim.y = s_im[i1];
          cre[t] = __builtin_amdgcn_wmma_f32_16x16x4_f32(
              false, a, false, bre, (short)0, cre[t], false, false);
          cim[t] = __builtin_amdgcn_wmma_f32_16x16x4_f32(
              false, a, false, bim, (short)0, cim[t], false, false);
        }
      }
      __syncthreads();  // all LDS reads done -> safe in-place update

#pragma unroll
      for (int t = 0; t < kTilesPerWave; ++t) {
        const int c = (wv * kTilesPerWave + t) * 16 + n;
#pragma unroll
        for (int j = 0; j < 8; ++j) {
          const int m   = j + 8 * half;
          const int idx = state_index(g, m, c);
          s_re[idx] = cre[t][j];
          s_im[idx] = cim[t][j];
        }
      }
      __syncthreads();
    }

    // ---- RZ diagonal fused with composed CNOT-chain permutation ----
    float tz[kNQ], sumt = 0.0f;
#pragma unroll
    for (int p = 0; p < kNQ; ++p) {
      tz[p] = pL[kNQ + (kNQ - 1 - p)];  // RZ param for flat bit p
      sumt += tz[p];
    }
    float gre[kAPT], gim[kAPT];
#pragma unroll
    for (int r = 0; r < kAPT; ++r) {
      const int k = r * kTPB + tid;
      // Chain: CNOT(q_i->q_{i+1}) i=0..10, then CNOT(q11->q0).
      // In flat-bit space (bit p == qubit 11-p), inverted composition:
      int j = k;
      j ^= (j & 1) << 11;  // last gate: ctrl bit0 -> tgt bit11
#pragma unroll
      for (int cb = 1; cb < kNQ; ++cb)  // gates ctrl bit cb -> tgt bit cb-1
        j ^= ((j >> cb) & 1) << (cb - 1);
      float phi = -0.5f * sumt;  // exp(i*phi), phi = sum_p tz_p*(bit_p - 1/2)
#pragma unroll
      for (int p = 0; p < kNQ; ++p) phi += ((j >> p) & 1) ? tz[p] : 0.0f;
      float sp, cp;
      __sincosf(phi, &sp, &cp);
      const float sr = s_re[j], si = s_im[j];
      gre[r] = sr * cp - si * sp;
      gim[r] = sr * sp + si * cp;
    }
    __syncthreads();
#pragma unroll
    for (int r = 0; r < kAPT; ++r) {
      const int k = r * kTPB + tid;
      s_re[k] = gre[r];
      s_im[k] = gim[r];
    }
    __syncthreads();
  }

  // ---- <sum_i Z_i>, then Linear(1,1) + sigmoid ----
  float partial = 0.0f;
#pragma unroll
  for (int r = 0; r < kAPT; ++r) {
    const int k = r * kTPB + tid;
    const float pr = s_re[k] * s_re[k] + s_im[k] * s_im[k];
    partial += pr * (float)(kNQ - 2 * __popc(k));
  }
  s_gate[tid] = partial;
  __syncthreads();
#pragma unroll
  for (int stride = kTPB / 2; stride > 0; stride >>= 1) {
    if (tid < stride) s_gate[tid] += s_gate[tid + stride];
    __syncthreads();
  }
  if (tid == 0) {
    const float v = (s_gate[0] / (float)kNQ) * lin_w[0] + lin_b[0];
    out[b] = 1.0f / (1.0f + __expf(-v));
  }
}

extern "C" void kernel_launch(void* const* d_in, const int* in_sizes, int n_in,
                              void* d_out, int out_size, void* d_ws,
                              size_t ws_size, hipStream_t stream) {
  (void)in_sizes; (void)n_in; (void)d_ws; (void)ws_size;
  const float* x      = (const float*)d_in[0];  // (B, 12)
  const float* params = (const float*)d_in[1];  // (144,)
  const float* lw     = (const float*)d_in[2];  // (1,1)
  const float* lb     = (const float*)d_in[3];  // (1,)
  float* out          = (float*)d_out;          // (B, 1) f32

  qnn_vqc_kernel<<<dim3(out_size), dim3(kTPB), 0, stream>>>(x, params, lw, lb,
                                                            out);
}